// PositionalSAE_11768210391682
// MI455X (gfx1250) — compile-verified
//
#include <hip/hip_runtime.h>

typedef float v2f __attribute__((ext_vector_type(2)));
typedef float v4f __attribute__((ext_vector_type(4)));
typedef float v8f __attribute__((ext_vector_type(8)));
typedef unsigned int v4u __attribute__((ext_vector_type(4)));
typedef int v4i __attribute__((ext_vector_type(4)));
typedef int v8i __attribute__((ext_vector_type(8)));

#if defined(__has_builtin)
#if __has_builtin(__builtin_amdgcn_tensor_load_to_lds) && \
    __has_builtin(__builtin_amdgcn_s_wait_tensorcnt)
#define USE_TDM 1
#endif
#endif
#ifndef USE_TDM
#define USE_TDM 0
#endif

namespace {
constexpr int kDin     = 2048;   // d_in
constexpr int kChunk   = 4096;   // d_sae / n_ranges
constexpr int kNR      = 4;      // ranges
constexpr int kRB      = 32;     // batch rows per range
constexpr int kKC      = 64;     // K chunk staged in LDS
constexpr int kNT      = 32;     // N tile per workgroup
constexpr int kThreads = 128;    // 4 waves (wave32)
constexpr float kEps   = 1e-5f;

// Padded LDS tile layouts (pad via TDM D# pad_interval/pad_amount):
// A tile: 32 rows x 64 dwords, pad 4 dwords per 64  -> row stride 68 dwords
//   fragment ds_load_b64: lane bank advance (2*68 banks-of-8B) -> conflict-free
// B tile: 64 rows x 32 dwords, pad 8 dwords per 32  -> row stride 40 dwords
//   fragment b32 pair: half-waves land on disjoint bank sets
constexpr int kALd    = 68;
constexpr int kBLd    = 40;
constexpr int kABytes = kRB * kALd * 4;              // 8704
constexpr int kBBytes = kKC * kBLd * 4;              // 10240
constexpr int kLdsBytes = 2 * kABytes + 2 * kBBytes; // 37888
}

#if USE_TDM
// ---------------------------------------------------------------------------
// Build a 2D Tensor-DMA descriptor (ISA 8.3-8.6) and issue TENSOR_LOAD_TO_LDS.
//   tile_d0 = contiguous elements per row, tile_d1 = rows,
//   row_stride in elements (48-bit), f32 data (data_size code 2).
//   pad codes: interval dwords = 2<<code ; amount dwords = code+1.
// ---------------------------------------------------------------------------
__device__ __forceinline__ void tdm_load_2d(unsigned int lds_off,
                                            const float* gptr,
                                            unsigned int tile_d0,
                                            unsigned int tile_d1,
                                            unsigned long long row_stride,
                                            unsigned int pad_int_code,
                                            unsigned int pad_amt_code)
{
  const unsigned long long ga = (unsigned long long)(const void*)gptr;
  v4u g0;
  g0[0] = 1u;                                            // count=1 (user D#)
  g0[1] = lds_off;                                       // lds_addr (bytes)
  g0[2] = (unsigned int)ga;                              // global_addr[31:0]
  g0[3] = (unsigned int)((ga >> 32) & 0x01ffffffull)     // global_addr[56:32]
          | (2u << 30);                                  // type = 2 ("image")
  v8i g1;
  g1[0] = (int)((2u << 16)                               // data_size = 4B
                | (1u << 20)                             // pad_enable
                | (pad_int_code << 22)
                | (pad_amt_code << 25));
  g1[1] = (int)(tile_d0 << 16);                          // tensor_dim0[15:0]
  g1[2] = (int)(tile_d1 << 16);                          // dim0 hi=0 | tensor_dim1[15:0]
  g1[3] = (int)(tile_d0 << 16);                          // dim1 hi=0 | tile_dim0
  g1[4] = (int)(tile_d1 & 0xffffu);                      // tile_dim1 | tile_dim2=0
  g1[5] = (int)(row_stride & 0xffffffffull);             // dim0_stride[31:0]
  g1[6] = (int)((row_stride >> 32) & 0xffffull);         // dim0_stride[47:32]
  g1[7] = 0;                                             // dim1_stride unused (2D)
  const v4i z4 = {0, 0, 0, 0};
#if __clang_major__ >= 23
  const v8i z8 = {0, 0, 0, 0, 0, 0, 0, 0};
  __builtin_amdgcn_tensor_load_to_lds(g0, g1, z4, z4, z8, 0);
#else
  __builtin_amdgcn_tensor_load_to_lds(g0, g1, z4, z4, 0);
#endif
}
#endif  // USE_TDM

// ---------------------------------------------------------------------------
// Shared GEMM core: C(32x32 tile) += A(32xKTOT) * B(KTOT x kNT slice).
// 4 waves, one 16x16 V_WMMA_F32_16X16X4_F32 accumulator chain each.
// Double-buffered LDS chunks fed by TDM (wave 0 only) or sync fallback.
// ---------------------------------------------------------------------------
template <int KTOT, int LDA, int LDB>
__device__ __forceinline__ v8f gemm_tile_core(const float* __restrict__ Ag,
                                              const float* __restrict__ Bg)
{
  const int tid  = threadIdx.x;
  const int lane = tid & 31;
  const int wave = tid >> 5;
  const int mt   = wave >> 1;          // M half
  const int nc   = (wave & 1) * 16;    // N sub-column
  const int half = lane >> 4;          // K-pair select (ISA f32 A/B layout)
  const int l    = lane & 15;

  __shared__ __align__(128) unsigned char smem[kLdsBytes];
  // NOTE: compute buffer pointers dynamically; building pointer ARRAYS from a
  // __shared__ base creates addrspacecast static initializers that ld.lld
  // rejects for gfx1250.
  unsigned char* sb = smem;
  auto Aptr = [&](int b) -> float* {
    return (float*)(sb + (size_t)b * kABytes);
  };
  auto Bptr = [&](int b) -> float* {
    return (float*)(sb + 2 * kABytes + (size_t)b * kBBytes);
  };

  v8f acc = {0.f, 0.f, 0.f, 0.f, 0.f, 0.f, 0.f, 0.f};

  auto stage = [&](int kb, int buf) {
#if USE_TDM
    if (wave == 0) {
      const unsigned int ab = (unsigned int)(unsigned long long)(void*)Aptr(buf);
      const unsigned int bb = (unsigned int)(unsigned long long)(void*)Bptr(buf);
      // A: 32 rows x 64 contiguous K ; pad 4 dwords each 64 -> stride 68
      tdm_load_2d(ab, Ag + kb, kKC, kRB, (unsigned long long)LDA, 5u, 3u);
      // B: 64 K-rows x 32 contiguous N ; pad 8 dwords each 32 -> stride 40
      tdm_load_2d(bb, Bg + (size_t)kb * LDB, kNT, kKC,
                  (unsigned long long)LDB, 4u, 7u);
    }
#else
    float* Al = Aptr(buf);
    float* Bl = Bptr(buf);
    #pragma unroll
    for (int it = 0; it < (kKC * kRB / 2) / kThreads; ++it) {   // 8 iters
      int i = it * kThreads + tid;
      int p = i & 31;                  // K pair
      int m = i >> 5;                  // row
      v2f a = *(const v2f*)(Ag + (size_t)m * LDA + kb + 2 * p);
      *(v2f*)(Al + m * kALd + 2 * p) = a;
    }
    #pragma unroll
    for (int it = 0; it < (kKC * kNT / 4) / kThreads; ++it) {   // 4 iters
      int i  = it * kThreads + tid;
      int ng = i & 7;
      int k  = i >> 3;
      v4f b = *(const v4f*)(Bg + (size_t)(kb + k) * LDB + ng * 4);
      *(v4f*)(Bl + k * kBLd + ng * 4) = b;
    }
#endif
  };

  auto fence = [&]() {
#if USE_TDM
    if (wave == 0) __builtin_amdgcn_s_wait_tensorcnt(0);
#endif
  };

  stage(0, 0);
  fence();
  __syncthreads();

  constexpr int nch = KTOT / kKC;
  for (int c = 0; c < nch; ++c) {
    const int cur = c & 1;
    if (c + 1 < nch) stage((c + 1) * kKC, cur ^ 1);   // DMA next chunk

    const float* A = Aptr(cur) + (mt * 16 + l) * kALd + 2 * half;
    const float* B = Bptr(cur) + (2 * half) * kBLd + nc + l;
    #pragma unroll
    for (int k0 = 0; k0 < kKC; k0 += 4) {
      v2f af = *(const v2f*)(A + k0);                  // ds_load_b64
      v2f bf;
      bf.x = B[(size_t)k0 * kBLd];                     // ds 2-addr b32 pair
      bf.y = B[(size_t)k0 * kBLd + kBLd];
      acc = __builtin_amdgcn_wmma_f32_16x16x4_f32(false, af, false, bf,
                                                  (short)0, acc, false, false);
    }
    fence();
    __syncthreads();
  }
  return acc;
}

// ---------------------------------------------------------------------------
// Prologue: x = acts + pos_emb[range] - b_dec  (pure-copy GEMM input for TDM)
// ---------------------------------------------------------------------------
__global__ __launch_bounds__(256)
void sae_prep_x_kernel(const float* __restrict__ acts,
                       const float* __restrict__ pos,
                       const float* __restrict__ b_dec,
                       float* __restrict__ x)
{
  const int i   = blockIdx.x * 256 + threadIdx.x;   // float4 index, 65536 total
  const int row = i >> 9;                           // 512 float4 per row
  const int c4  = i & 511;
  const int r   = row >> 5;
  v4f a  = *(const v4f*)(acts + (size_t)i * 4);
  v4f pe = *(const v4f*)(pos + (size_t)(r * 512 + c4) * 4);
  v4f bd = *(const v4f*)(b_dec + (size_t)c4 * 4);
  v4f o;
  o.x = a.x + pe.x - bd.x;
  o.y = a.y + pe.y - bd.y;
  o.z = a.z + pe.z - bd.z;
  o.w = a.w + pe.w - bd.w;
  *(v4f*)(x + (size_t)i * 4) = o;
}

// ---------------------------------------------------------------------------
// Encode: pre[r*32+m][col] = (x[r] @ W_enc[r])[m][col] + b_enc[r*4096+col]
// ---------------------------------------------------------------------------
__global__ __launch_bounds__(kThreads)
void sae_encode_kernel(const float* __restrict__ x,
                       const float* __restrict__ Wenc,
                       const float* __restrict__ b_enc,
                       float* __restrict__ pre)
{
  constexpr int ntiles = kChunk / kNT;              // 128
  const int r  = blockIdx.x / ntiles;
  const int nt = blockIdx.x % ntiles;
  const int n0 = nt * kNT;

  const float* Ag = x + (size_t)(r * kRB) * kDin;
  const float* Bg = Wenc + (size_t)r * kDin * kChunk + n0;
  v8f acc = gemm_tile_core<kDin, kDin, kChunk>(Ag, Bg);

  const int lane = threadIdx.x & 31;
  const int wave = threadIdx.x >> 5;
  const int mt = wave >> 1, nc = (wave & 1) * 16;
  const int half = lane >> 4, l = lane & 15;
  const int col = n0 + nc + l;
  const float be = b_enc[(size_t)r * kChunk + col];
  #pragma unroll
  for (int j = 0; j < 8; ++j) {                     // D: VGPR j -> M=j+half*8
    const int mrow = r * kRB + mt * 16 + half * 8 + j;
    pre[(size_t)mrow * kChunk + col] = acc[j] + be;
  }
}

// ---------------------------------------------------------------------------
// Per-row LayerNorm(4096) + gamma/beta + ReLU, in place (thread-owned elems).
// ---------------------------------------------------------------------------
__global__ __launch_bounds__(256)
void sae_ln_relu_kernel(float* __restrict__ pre,
                        const float* __restrict__ gamma,
                        const float* __restrict__ beta)
{
  const int row = blockIdx.x;                       // 0..127
  const int r   = row / kRB;
  float* p      = pre + (size_t)row * kChunk;
  const int t   = threadIdx.x;

  float v[16];
  float s = 0.f;
  #pragma unroll
  for (int i = 0; i < 16; ++i) { v[i] = p[t + i * 256]; s += v[i]; }

  __shared__ float red[256];
  red[t] = s;
  __syncthreads();
  #pragma unroll
  for (int off = 128; off > 0; off >>= 1) {
    if (t < off) red[t] += red[t + off];
    __syncthreads();
  }
  const float mu = red[0] * (1.f / (float)kChunk);
  __syncthreads();

  float sq = 0.f;
  #pragma unroll
  for (int i = 0; i < 16; ++i) { float d = v[i] - mu; sq += d * d; }
  red[t] = sq;
  __syncthreads();
  #pragma unroll
  for (int off = 128; off > 0; off >>= 1) {
    if (t < off) red[t] += red[t + off];
    __syncthreads();
  }
  const float rs = rsqrtf(red[0] * (1.f / (float)kChunk) + kEps);

  #pragma unroll
  for (int i = 0; i < 16; ++i) {
    const int c = t + i * 256;
    const float g = gamma[(size_t)r * kChunk + c];
    const float b = beta[(size_t)r * kChunk + c];
    const float o = (v[i] - mu) * rs * g + b;
    p[c] = fmaxf(o, 0.f);                           // feat, in place
  }
}

// ---------------------------------------------------------------------------
// Decode (block-diagonal): out[r*32+m][col] =
//     (feat[r] @ W_dec[r*4096:(r+1)*4096])[m][col] + b_dec[col]
// ---------------------------------------------------------------------------
__global__ __launch_bounds__(kThreads)
void sae_decode_kernel(const float* __restrict__ feat,
                       const float* __restrict__ Wdec,
                       const float* __restrict__ b_dec,
                       float* __restrict__ out)
{
  constexpr int ntiles = kDin / kNT;                // 64
  const int r  = blockIdx.x / ntiles;
  const int nt = blockIdx.x % ntiles;
  const int n0 = nt * kNT;

  const float* Ag = feat + (size_t)(r * kRB) * kChunk;
  const float* Bg = Wdec + (size_t)(r * kChunk) * kDin + n0;
  v8f acc = gemm_tile_core<kChunk, kChunk, kDin>(Ag, Bg);

  const int lane = threadIdx.x & 31;
  const int wave = threadIdx.x >> 5;
  const int mt = wave >> 1, nc = (wave & 1) * 16;
  const int half = lane >> 4, l = lane & 15;
  const int col = n0 + nc + l;
  const float bd = b_dec[col];
  #pragma unroll
  for (int j = 0; j < 8; ++j) {
    const int mrow = r * kRB + mt * 16 + half * 8 + j;
    out[(size_t)mrow * kDin + col] = acc[j] + bd;
  }
}

// ---------------------------------------------------------------------------
extern "C" void kernel_launch(void* const* d_in, const int* in_sizes, int n_in,
                              void* d_out, int out_size, void* d_ws, size_t ws_size,
                              hipStream_t stream)
{
  const float* acts  = (const float*)d_in[0];   // [128][2048]
  const float* Wenc  = (const float*)d_in[1];   // [4][2048][4096]
  const float* b_enc = (const float*)d_in[2];   // [16384]
  const float* b_dec = (const float*)d_in[3];   // [2048]
  const float* pos   = (const float*)d_in[4];   // [4][2048]
  const float* gamma = (const float*)d_in[5];   // [16384]
  const float* beta  = (const float*)d_in[6];   // [16384]
  const float* Wdec  = (const float*)d_in[7];   // [16384][2048]
  float* out = (float*)d_out;                   // [128][2048]

  float* x   = (float*)d_ws;                    // 128*2048 floats (1 MB)
  float* pre = x + (size_t)128 * kDin;          // 128*4096 floats (2 MB)

  sae_prep_x_kernel<<<dim3(256), dim3(256), 0, stream>>>(acts, pos, b_dec, x);
  sae_encode_kernel<<<dim3(kNR * (kChunk / kNT)), dim3(kThreads), 0, stream>>>(
      x, Wenc, b_enc, pre);
  sae_ln_relu_kernel<<<dim3(kNR * kRB), dim3(256), 0, stream>>>(pre, gamma, beta);
  sae_decode_kernel<<<dim3(kNR * (kDin / kNT)), dim3(kThreads), 0, stream>>>(
      pre, Wdec, b_dec, out);
}